// GeometricHyperConnections_30391188587034
// MI455X (gfx1250) — compile-verified
//
#include <hip/hip_runtime.h>
#include <math.h>

typedef float v2f __attribute__((ext_vector_type(2)));
typedef float v8f __attribute__((ext_vector_type(8)));
typedef unsigned int u32x4 __attribute__((ext_vector_type(4)));
typedef int i32x8 __attribute__((ext_vector_type(8)));
typedef int i32x4 __attribute__((ext_vector_type(4)));

#define D_MODEL 2048
#define S_STREAMS 4
#define KDIM 4
#define BLOCK 256
#define TAU 0.05f
#define SINK_ITERS 10
#define RMS_EPS 1e-6f

#if defined(__HIP_DEVICE_COMPILE__) && __has_builtin(__builtin_amdgcn_tensor_load_to_lds)
#define USE_TDM 1
#else
#define USE_TDM 0
#endif

__device__ __forceinline__ float lse4(float a, float b, float c, float d) {
  float m = fmaxf(fmaxf(a, b), fmaxf(c, d));
  return m + logf(expf(a - m) + expf(b - m) + expf(c - m) + expf(d - m));
}

#if USE_TDM
// Issue one TDM 2D tile load: `rows` rows of `row_len` f32, rows strided by
// `row_stride_elts` f32 in global memory, packed contiguously into LDS.
// D# layout per CDNA5 ISA ch.8 (group0: control/lds/global addr, group1: dims).
// clang-23 amdgpu-toolchain builtin: 6 args (g0, g1, g2, g3, g4, cpol).
__device__ __forceinline__ void tdm_load_2d(unsigned long long gaddr, unsigned lds_addr,
                                            unsigned rows, unsigned long long row_stride_elts,
                                            unsigned row_len) {
  u32x4 g0;
  g0[0] = 1u;                                        // count=1, user descriptor
  g0[1] = lds_addr;                                  // lds_addr (bytes)
  g0[2] = (unsigned)(gaddr & 0xFFFFFFFFu);           // global_addr[31:0]
  g0[3] = (unsigned)(gaddr >> 32) | (2u << 30);      // global_addr[56:32] | type=2
  i32x8 g1;
  g1[0] = (int)(2u << 16);                           // data_size=2 (4 bytes), no mask/flags
  g1[1] = (int)((row_len & 0xFFFFu) << 16);          // tensor_dim0[15:0] @ bits[63:48]
  g1[2] = (int)(((row_len >> 16) & 0xFFFFu)          // tensor_dim0[31:16]
              | ((rows & 0xFFFFu) << 16));           // tensor_dim1[15:0]
  g1[3] = (int)(((rows >> 16) & 0xFFFFu)             // tensor_dim1[31:16]
              | ((row_len & 0xFFFFu) << 16));        // tile_dim0
  g1[4] = (int)(rows & 0xFFFFu);                     // tile_dim1 (tile_dim2=0)
  g1[5] = (int)(unsigned)(row_stride_elts & 0xFFFFFFFFu);        // dim0_stride[31:0]
  g1[6] = (int)(unsigned)((row_stride_elts >> 32) & 0xFFFFu);    // dim0_stride[47:32]
  g1[7] = 0;                                         // dim1_stride unused (2D)
  i32x4 z4 = {0, 0, 0, 0};                           // groups 2/3 unused (<=3D)
  i32x8 z8 = {0, 0, 0, 0, 0, 0, 0, 0};               // extra group (clang-23 form)
  __builtin_amdgcn_tensor_load_to_lds(g0, g1, z4, z4, z8, 0);
}
#endif

__global__ __launch_bounds__(BLOCK) void ghc_kernel(
    const float* __restrict__ residuals,     // (b*s, T, d)
    const float* __restrict__ branch_output, // (b, T, d)
    const float* __restrict__ rms_weight,    // (d)
    const float* __restrict__ proj_W,        // (k, d)
    const float* __restrict__ log_sigma,     // scalar
    const float* __restrict__ pre_logits,    // (s)
    const float* __restrict__ post_logits,   // (s)
    float* __restrict__ branch_input,        // (b, T, d)
    float* __restrict__ out,                 // (b*s, T, d)
    int T)
{
  __shared__ float lds_r[S_STREAMS][D_MODEL]; // 32 KB staged residual streams
  __shared__ float lds_bo[D_MODEL];           // 8 KB branch_output row
  __shared__ float red[8][5];                 // per-wave partials
  __shared__ float sh_stats[S_STREAMS][5];    // {ssq, dot0..3} per stream
  __shared__ float sh_A[16][S_STREAMS];       // WMMA A: rows 0-3 = H^T, row 4 = H_pre
  __shared__ float sh_hpost[S_STREAMS];

  const int tid  = threadIdx.x;
  const int blk  = blockIdx.x;
  const int bIdx = blk / T;
  const int t    = blk - bIdx * T;

  // ---- Phase 1: stage r (4 strided rows) and branch_output row into LDS ----
#if USE_TDM
  if (tid < 32) {   // wave 0 only: TDM is a wave-level DMA op
    unsigned long long r_gaddr = (unsigned long long)(uintptr_t)(residuals +
        (((size_t)(bIdx * S_STREAMS)) * (size_t)T + (size_t)t) * (size_t)D_MODEL);
    tdm_load_2d(r_gaddr, (unsigned)(uintptr_t)&lds_r[0][0],
                S_STREAMS, (unsigned long long)T * (unsigned long long)D_MODEL, D_MODEL);
    unsigned long long b_gaddr = (unsigned long long)(uintptr_t)(branch_output +
        ((size_t)bIdx * (size_t)T + (size_t)t) * (size_t)D_MODEL);
    tdm_load_2d(b_gaddr, (unsigned)(uintptr_t)&lds_bo[0],
                1, (unsigned long long)D_MODEL, D_MODEL);
#if __has_builtin(__builtin_amdgcn_s_wait_tensorcnt)
    __builtin_amdgcn_s_wait_tensorcnt(0);
#else
    asm volatile("s_wait_tensorcnt 0x0" ::: "memory");
#endif
  }
#else
  {
    const int sIdx0 = tid >> 6;
    const int li0   = tid & 63;
    const float4* rsrc = (const float4*)(residuals +
        (((size_t)(bIdx * S_STREAMS + sIdx0)) * (size_t)T + (size_t)t) * (size_t)D_MODEL);
#pragma unroll
    for (int j = 0; j < (D_MODEL / 4) / 64; ++j) {
      int c4 = li0 + j * 64;
      ((float4*)(&lds_r[sIdx0][0]))[c4] = rsrc[c4];
    }
    const float4* bsrc = (const float4*)(branch_output +
        ((size_t)bIdx * (size_t)T + (size_t)t) * (size_t)D_MODEL);
#pragma unroll
    for (int j = 0; j < (D_MODEL / 4) / BLOCK; ++j) {
      int c4 = tid + j * BLOCK;
      ((float4*)lds_bo)[c4] = bsrc[c4];
    }
  }
#endif
  __syncthreads();

  // ---- Phase 2: per-stream sum-of-squares + projection dots (r from LDS) ----
  const int sIdx = tid >> 6;     // 64 threads per stream
  const int li   = tid & 63;
  const float4* wsrc  = (const float4*)rms_weight;
  const float4* p0src = (const float4*)(proj_W + 0 * D_MODEL);
  const float4* p1src = (const float4*)(proj_W + 1 * D_MODEL);
  const float4* p2src = (const float4*)(proj_W + 2 * D_MODEL);
  const float4* p3src = (const float4*)(proj_W + 3 * D_MODEL);

  float ssq = 0.f, dk0 = 0.f, dk1 = 0.f, dk2 = 0.f, dk3 = 0.f;
#pragma unroll
  for (int j = 0; j < (D_MODEL / 4) / 64; ++j) {   // 8 iterations
    int c4 = li + j * 64;
    float4 rv = ((const float4*)(&lds_r[sIdx][0]))[c4];
    ssq += rv.x * rv.x + rv.y * rv.y + rv.z * rv.z + rv.w * rv.w;
    float4 wv = wsrc[c4];
    float rwx = rv.x * wv.x, rwy = rv.y * wv.y, rwz = rv.z * wv.z, rww = rv.w * wv.w;
    float4 p;
    p = p0src[c4]; dk0 += rwx * p.x + rwy * p.y + rwz * p.z + rww * p.w;
    p = p1src[c4]; dk1 += rwx * p.x + rwy * p.y + rwz * p.z + rww * p.w;
    p = p2src[c4]; dk2 += rwx * p.x + rwy * p.y + rwz * p.z + rww * p.w;
    p = p3src[c4]; dk3 += rwx * p.x + rwy * p.y + rwz * p.z + rww * p.w;
  }

  // wave32 shuffle reduction, then cross-wave via LDS
  float acc[5] = { ssq, dk0, dk1, dk2, dk3 };
#pragma unroll
  for (int off = 16; off >= 1; off >>= 1) {
#pragma unroll
    for (int q = 0; q < 5; ++q) acc[q] += __shfl_xor(acc[q], off, 32);
  }
  const int wave = tid >> 5;
  const int lane = tid & 31;
  if (lane == 0) {
#pragma unroll
    for (int q = 0; q < 5; ++q) red[wave][q] = acc[q];
  }
  __syncthreads();
  if (tid < S_STREAMS) {  // two waves per stream
#pragma unroll
    for (int q = 0; q < 5; ++q)
      sh_stats[tid][q] = red[2 * tid][q] + red[2 * tid + 1][q];
  }
  __syncthreads();

  // ---- Phase 3: Sinkhorn 4x4 on one thread (negligible vs memory) ----
  if (tid == 0) {
    float coords[S_STREAMS][KDIM];
#pragma unroll
    for (int s = 0; s < S_STREAMS; ++s) {
      float inv = rsqrtf(sh_stats[s][0] * (1.0f / (float)D_MODEL) + RMS_EPS);
#pragma unroll
      for (int k = 0; k < KDIM; ++k) coords[s][k] = sh_stats[s][1 + k] * inv;
    }
    float sigma_sq = expf(2.0f * log_sigma[0]);
    float zscale = -1.0f / (2.0f * sigma_sq * TAU);
    float Z[S_STREAMS][S_STREAMS];
#pragma unroll
    for (int i = 0; i < S_STREAMS; ++i)
#pragma unroll
      for (int j = 0; j < S_STREAMS; ++j) {
        float ds = 0.f;
#pragma unroll
        for (int k = 0; k < KDIM; ++k) {
          float df = coords[i][k] - coords[j][k];
          ds += df * df;
        }
        Z[i][j] = ds * zscale;
      }
    float uu[S_STREAMS] = {0.f, 0.f, 0.f, 0.f};
    float vv[S_STREAMS] = {0.f, 0.f, 0.f, 0.f};
    const float lm = -logf((float)S_STREAMS);
    for (int it = 0; it < SINK_ITERS; ++it) {
#pragma unroll
      for (int i = 0; i < S_STREAMS; ++i)
        uu[i] = lm - lse4(Z[i][0] + vv[0], Z[i][1] + vv[1], Z[i][2] + vv[2], Z[i][3] + vv[3]);
#pragma unroll
      for (int j = 0; j < S_STREAMS; ++j)
        vv[j] = lm - lse4(Z[0][j] + uu[0], Z[1][j] + uu[1], Z[2][j] + uu[2], Z[3][j] + uu[3]);
    }
    // A matrix: Amat[u][s] = H[s][u]; row 4 = softmax(pre_logits); rows 5..15 = 0
#pragma unroll
    for (int r = 0; r < 16; ++r)
#pragma unroll
      for (int c = 0; c < S_STREAMS; ++c) sh_A[r][c] = 0.f;
#pragma unroll
    for (int i = 0; i < S_STREAMS; ++i)
#pragma unroll
      for (int j = 0; j < S_STREAMS; ++j)
        sh_A[j][i] = expf(Z[i][j] + uu[i] + vv[j]) * (float)S_STREAMS;

    float l0 = pre_logits[0], l1 = pre_logits[1], l2 = pre_logits[2], l3 = pre_logits[3];
    float m = fmaxf(fmaxf(l0, l1), fmaxf(l2, l3));
    float e0 = expf(l0 - m), e1 = expf(l1 - m), e2 = expf(l2 - m), e3 = expf(l3 - m);
    float inv = 1.0f / (e0 + e1 + e2 + e3);
    sh_A[4][0] = e0 * inv; sh_A[4][1] = e1 * inv; sh_A[4][2] = e2 * inv; sh_A[4][3] = e3 * inv;

    l0 = post_logits[0]; l1 = post_logits[1]; l2 = post_logits[2]; l3 = post_logits[3];
    m = fmaxf(fmaxf(l0, l1), fmaxf(l2, l3));
    e0 = expf(l0 - m); e1 = expf(l1 - m); e2 = expf(l2 - m); e3 = expf(l3 - m);
    inv = 1.0f / (e0 + e1 + e2 + e3);
    sh_hpost[0] = e0 * inv; sh_hpost[1] = e1 * inv; sh_hpost[2] = e2 * inv; sh_hpost[3] = e3 * inv;
  }
  __syncthreads();

  // ---- Phase 4: mixing via V_WMMA_F32_16X16X4_F32 ----
  // A (16x4): lanes 0-15 hold K=0(V0),K=1(V1); lanes 16-31 hold K=2,K=3 for M=lane%16
  const int Mrow = lane & 15;
  const int kh   = (lane < 16) ? 0 : 2;
  v2f A;
  A.x = sh_A[Mrow][kh];
  A.y = sh_A[Mrow][kh + 1];

  const float hp0 = sh_hpost[0], hp1 = sh_hpost[1], hp2 = sh_hpost[2], hp3 = sh_hpost[3];
  const size_t rowStride = (size_t)T * (size_t)D_MODEL;
  const size_t outRow0 = (((size_t)(bIdx * S_STREAMS)) * (size_t)T + (size_t)t) * (size_t)D_MODEL;
  const size_t biRow   = ((size_t)bIdx * (size_t)T + (size_t)t) * (size_t)D_MODEL;
  const bool lo = (lane < 16);

#pragma unroll 2
  for (int tile = wave; tile < D_MODEL / 16; tile += BLOCK / 32) {
    int col = tile * 16 + Mrow;
    // B (4x16): lanes 0-15 rows K=0/1, lanes 16-31 rows K=2/3, N=lane%16
    v2f B;
    B.x = lds_r[kh][col];
    B.y = lds_r[kh + 1][col];
    float bov = lds_bo[col];
    // C (16x16 f32): VGPR v holds M=v (lanes 0-15) / M=v+8 (lanes 16-31)
    v8f C;
    C[0] = lo ? bov * hp0 : 0.f;
    C[1] = lo ? bov * hp1 : 0.f;
    C[2] = lo ? bov * hp2 : 0.f;
    C[3] = lo ? bov * hp3 : 0.f;
    C[4] = 0.f; C[5] = 0.f; C[6] = 0.f; C[7] = 0.f;

    v8f Dv = __builtin_amdgcn_wmma_f32_16x16x4_f32(
        false, A, false, B, (short)0, C, false, false);

    if (lo) {
      out[outRow0 + 0 * rowStride + col] = Dv[0];  // 64B-aligned 64B chunks
      out[outRow0 + 1 * rowStride + col] = Dv[1];
      out[outRow0 + 2 * rowStride + col] = Dv[2];
      out[outRow0 + 3 * rowStride + col] = Dv[3];
      branch_input[biRow + col] = Dv[4];
    }
  }
}

extern "C" void kernel_launch(void* const* d_in, const int* in_sizes, int n_in,
                              void* d_out, int out_size, void* d_ws, size_t ws_size,
                              hipStream_t stream) {
  (void)n_in; (void)d_ws; (void)ws_size; (void)out_size;
  const float* residuals     = (const float*)d_in[0];
  const float* branch_output = (const float*)d_in[1];
  const float* rms_weight    = (const float*)d_in[2];
  const float* proj_W        = (const float*)d_in[3];
  const float* log_sigma     = (const float*)d_in[4];
  const float* pre_logits    = (const float*)d_in[5];
  const float* post_logits   = (const float*)d_in[6];

  const int dmodel = in_sizes[2];                 // 2048
  const long bT = (long)in_sizes[1] / dmodel;     // b*T = 8192
  const int T = 2048;                             // reference T

  float* branch_input = (float*)d_out;                        // (b, T, d) first
  float* outp = branch_input + (size_t)bT * (size_t)dmodel;   // then (b*s, T, d)

  dim3 grid((unsigned)bT);
  dim3 block(BLOCK);
  hipLaunchKernelGGL(ghc_kernel, grid, block, 0, stream,
                     residuals, branch_output, rms_weight, proj_W, log_sigma,
                     pre_logits, post_logits, branch_input, outp, T);
}